// EdgeInteractionNet_6210522710114
// MI455X (gfx1250) — compile-verified
//
#include <hip/hip_runtime.h>

typedef float v2f __attribute__((ext_vector_type(2)));
typedef float v8f __attribute__((ext_vector_type(8)));

#define N_NODES 10000
#define N_EDGES 320000
#define N_FEAT  128
#define N_BASIS 20
#define EPB     64     // edges per block in kernel 2 (4 M-tiles of 16)

// ---------------------------------------------------------------------------
// Kernel 1: message_nodepart = silu(X @ W1 + b1) @ W2 + b2
// One block = 16 nodes. 8 waves, each wave owns a 16-feature output slice.
// ---------------------------------------------------------------------------
__global__ __launch_bounds__(256) void node_mlp_kernel(
    const float* __restrict__ X,  const float* __restrict__ W1,
    const float* __restrict__ b1, const float* __restrict__ W2,
    const float* __restrict__ b2, float* __restrict__ nodepart)
{
    __shared__ float sX[16 * N_FEAT];   // 8 KB input tile
    __shared__ float sH[16 * N_FEAT];   // 8 KB hidden tile

    const int tid  = threadIdx.x;
    const int wave = tid >> 5;
    const int lane = tid & 31;
    const int base = blockIdx.x * 16;           // first node of tile

    // Coalesced copy of the contiguous 16x128 tile (512 float4, 2 per thread)
    {
        const float4* src = (const float4*)(X + (size_t)base * N_FEAT);
        float4* dst = (float4*)sX;
        dst[tid]       = src[tid];
        dst[tid + 256] = src[tid + 256];
    }
    __syncthreads();

    const int m     = lane & 15;   // A row
    const int kh    = lane >> 4;   // K half-select
    const int n     = lane & 15;   // B/C column
    const int nbase = wave * 16;   // this wave's feature slice

    // ---- stage 1: h = silu(X@W1 + b1) -------------------------------------
    v8f acc;
    #pragma unroll
    for (int r = 0; r < 8; ++r) acc[r] = b1[nbase + n];

    #pragma unroll 8
    for (int k0 = 0; k0 < N_FEAT; k0 += 4) {
        v2f a, b;
        a.x = sX[m * N_FEAT + k0     + 2 * kh];
        a.y = sX[m * N_FEAT + k0 + 1 + 2 * kh];
        b.x = W1[(k0     + 2 * kh) * N_FEAT + nbase + n];
        b.y = W1[(k0 + 1 + 2 * kh) * N_FEAT + nbase + n];
        acc = __builtin_amdgcn_wmma_f32_16x16x4_f32(
                  false, a, false, b, (short)0, acc, false, false);
    }

    // SiLU via fast reciprocal (v_exp_f32 + v_rcp_f32, no IEEE div slow path),
    // stage hidden tile in LDS per C-matrix layout (M = r + 8*kh)
    #pragma unroll
    for (int r = 0; r < 8; ++r) {
        float x = acc[r];
        float sig = __builtin_amdgcn_rcpf(1.0f + __expf(-x));
        sH[(r + 8 * kh) * N_FEAT + nbase + n] = x * sig;
    }
    __syncthreads();

    // ---- stage 2: out = h@W2 + b2 -----------------------------------------
    v8f acc2;
    #pragma unroll
    for (int r = 0; r < 8; ++r) acc2[r] = b2[nbase + n];

    #pragma unroll 8
    for (int k0 = 0; k0 < N_FEAT; k0 += 4) {
        v2f a, b;
        a.x = sH[m * N_FEAT + k0     + 2 * kh];
        a.y = sH[m * N_FEAT + k0 + 1 + 2 * kh];
        b.x = W2[(k0     + 2 * kh) * N_FEAT + nbase + n];
        b.y = W2[(k0 + 1 + 2 * kh) * N_FEAT + nbase + n];
        acc2 = __builtin_amdgcn_wmma_f32_16x16x4_f32(
                   false, a, false, b, (short)0, acc2, false, false);
    }

    #pragma unroll
    for (int r = 0; r < 8; ++r) {
        nodepart[(size_t)(base + r + 8 * kh) * N_FEAT + nbase + n] = acc2[r];
    }
}

// ---------------------------------------------------------------------------
// Kernel 2: out[e,f] = (dist[e,:] @ We)[f] * nodepart[src[e],f] * nodepart[dst[e],f]
// One block = 64 edges (4 M-tiles of 16). B fragments of We are loaded into
// registers ONCE per wave and reused across all 4 tiles (20 wmma per wave).
// ---------------------------------------------------------------------------
__global__ __launch_bounds__(256) void edge_fused_kernel(
    const float*     __restrict__ dist, const long long* __restrict__ eidx,
    const float*     __restrict__ We,   const float*     __restrict__ nodepart,
    float*           __restrict__ out)
{
    __shared__ float sD[EPB * N_BASIS];  // 5 KB dist tile
    __shared__ int   sSrc[EPB];
    __shared__ int   sDst[EPB];

    const int tid   = threadIdx.x;
    const int wave  = tid >> 5;
    const int lane  = tid & 31;
    const long long ebase = (long long)blockIdx.x * EPB;

    // dist tile is contiguous in memory: flat coalesced copy of 1280 floats
    #pragma unroll
    for (int i = 0; i < (EPB * N_BASIS) / 256; ++i)
        sD[tid + i * 256] = dist[ebase * N_BASIS + tid + i * 256];
    if (tid < EPB) {
        sSrc[tid] = (int)eidx[ebase + tid];                       // row 0: src
        sDst[tid] = (int)eidx[(long long)N_EDGES + ebase + tid];  // row 1: dst
    }
    __syncthreads();

    const int m     = lane & 15;
    const int kh    = lane >> 4;
    const int n     = lane & 15;
    const int nbase = wave * 16;

    // Load loop-invariant B fragments of We once (K = 20 = 5 steps of 4)
    v2f bfrag[5];
    #pragma unroll
    for (int kk = 0; kk < 5; ++kk) {
        const int k0 = kk * 4;
        bfrag[kk].x = We[(k0     + 2 * kh) * N_FEAT + nbase + n];
        bfrag[kk].y = We[(k0 + 1 + 2 * kh) * N_FEAT + nbase + n];
    }

    #pragma unroll
    for (int t = 0; t < EPB / 16; ++t) {
        const int mbase = t * 16;

        v8f acc = {};   // edge linear has no bias
        #pragma unroll
        for (int kk = 0; kk < 5; ++kk) {
            const int k0 = kk * 4;
            v2f a;
            a.x = sD[(mbase + m) * N_BASIS + k0     + 2 * kh];
            a.y = sD[(mbase + m) * N_BASIS + k0 + 1 + 2 * kh];
            acc = __builtin_amdgcn_wmma_f32_16x16x4_f32(
                      false, a, false, bfrag[kk], (short)0, acc, false, false);
        }

        // Fused dual gather (L2-resident nodepart) + triple product + store
        #pragma unroll
        for (int r = 0; r < 8; ++r) {
            const int mrow = mbase + r + 8 * kh;
            const int s = sSrc[mrow];
            const int d = sDst[mrow];
            const int f = nbase + n;
            const float sv = nodepart[(size_t)s * N_FEAT + f];
            const float dv = nodepart[(size_t)d * N_FEAT + f];
            out[(size_t)(ebase + mrow) * N_FEAT + f] = acc[r] * sv * dv;
        }
    }
}

// ---------------------------------------------------------------------------
extern "C" void kernel_launch(void* const* d_in, const int* in_sizes, int n_in,
                              void* d_out, int out_size, void* d_ws, size_t ws_size,
                              hipStream_t stream) {
    const float*     atom = (const float*)    d_in[0];  // [10000,128]
    const float*     dist = (const float*)    d_in[1];  // [320000,20]
    const long long* eidx = (const long long*)d_in[2];  // [2,320000] int64
    const float*     W1   = (const float*)    d_in[3];
    const float*     b1   = (const float*)    d_in[4];
    const float*     W2   = (const float*)    d_in[5];
    const float*     b2   = (const float*)    d_in[6];
    const float*     We   = (const float*)    d_in[7];
    float*           outp = (float*)d_out;

    // workspace: message_nodepart [10000,128] f32 = 5.12 MB (L2-resident)
    float* nodepart = (float*)d_ws;

    node_mlp_kernel<<<N_NODES / 16, 256, 0, stream>>>(atom, W1, b1, W2, b2, nodepart);
    edge_fused_kernel<<<N_EDGES / EPB, 256, 0, stream>>>(dist, eidx, We, nodepart, outp);
}